// GAT_28913719837096
// MI455X (gfx1250) — compile-verified
//
#include <hip/hip_runtime.h>

#define NNODES 50000
#define NEDGES 1600000
#define NTOT   (NEDGES + NNODES)   // edges + self loops
#define HEADS1 8
#define CH     16
#define HC1    128                 // HEADS1*CH
#define INCH   64
#define NB     256
#define OUTC   32

typedef __attribute__((ext_vector_type(2))) float v2f;
typedef __attribute__((ext_vector_type(8))) float v8f;

__device__ __forceinline__ float lrelu(float x) { return x > 0.f ? x : 0.2f * x; }
__device__ __forceinline__ float eluf(float x)  { return x > 0.f ? x : (__expf(x) - 1.f); }

// float atomic max via signed/unsigned int trick (monotone bit patterns)
__device__ __forceinline__ void atomicMaxF(float* addr, float v) {
  if (v >= 0.f) atomicMax((int*)addr, __float_as_int(v));
  else          atomicMin((unsigned int*)addr, __float_as_uint(v));
}

// ---------------------------------------------------------------------------
// Y[M,Nc] = X[M,K] @ W[K,Nc] + bias   — fp32 WMMA 16x16x4, one tile per wave32
// M must be a multiple of 16 (50000 = 3125*16), Nc a multiple of 16.
// A frag: lane(0..15)->M=lane,K={k0,k0+1}; lane(16..31)->M=lane-16,K={k0+2,k0+3}
// B frag: symmetric with N in place of M.  C/D: vgpr v -> M = v + 8*(lane>=16).
// ---------------------------------------------------------------------------
__global__ void gat_gemm_bias_wmma(const float* __restrict__ X,
                                   const float* __restrict__ W,
                                   const float* __restrict__ bias,
                                   float* __restrict__ Y,
                                   int M, int K, int Nc) {
  const int lane = threadIdx.x & 31;
  const int wave = blockIdx.x * (blockDim.x >> 5) + (threadIdx.x >> 5);
  const int ntiles = Nc >> 4;
  const int tm = (wave / ntiles) << 4;
  const int tn = (wave % ntiles) << 4;
  if (tm >= M) return;
  const int half = lane >> 4;       // 0: K pair {0,1}, 1: K pair {2,3}
  const int sub  = lane & 15;
  const int mrow = tm + sub;        // A-fragment row
  const int ncol = tn + sub;        // B-fragment / output column
  const float* __restrict__ xrow = X + (size_t)mrow * K;
  v8f acc = {};
  for (int k0 = 0; k0 < K; k0 += 4) {
    const int ka = k0 + 2 * half;
    v2f a, b;
    a.x = xrow[ka];
    a.y = xrow[ka + 1];
    b.x = W[(size_t)ka * Nc + ncol];
    b.y = W[(size_t)(ka + 1) * Nc + ncol];
    acc = __builtin_amdgcn_wmma_f32_16x16x4_f32(false, a, false, b,
                                                (short)0, acc, false, false);
  }
  const float bn = bias[ncol];
#pragma unroll
  for (int v = 0; v < 8; ++v) {
    const int m = tm + v + 8 * half;
    Y[(size_t)m * Nc + ncol] = acc[v] + bn;
  }
}

// ---------------------------------------------------------------------------
// fill constant (used for -inf init of segment-max buffers)
// ---------------------------------------------------------------------------
__global__ void gat_fill_f32(float* __restrict__ p, float v, int n) {
  const int i = blockIdx.x * blockDim.x + threadIdx.x;
  if (i < n) p[i] = v;
}

// ---------------------------------------------------------------------------
// Layer 1 edge passes (H=8, C=16): one wave32 per edge,
// lane -> (head = lane>>2, 4 channels = (lane&3)*4)
// ---------------------------------------------------------------------------
__device__ __forceinline__ void edge_endpoints(const int* __restrict__ ei, int e,
                                               int& src, int& dst) {
  if (e < NEDGES) { src = ei[e]; dst = ei[NEDGES + e]; }
  else            { src = dst = e - NEDGES; }
}

__global__ void gat_e1_logits_max(const int* __restrict__ ei,
                                  const float* __restrict__ xl,
                                  const float* __restrict__ xr,
                                  const float* __restrict__ att,
                                  float* __restrict__ logits,
                                  float* __restrict__ maxv) {
  const int lane = threadIdx.x & 31;
  const int e = blockIdx.x * (blockDim.x >> 5) + (threadIdx.x >> 5);
  if (e >= NTOT) return;
  int src, dst; edge_endpoints(ei, e, src, dst);
  const int h  = lane >> 2;
  const int hc = h * CH + (lane & 3) * 4;
  const float4 l4 = *(const float4*)(xl + (size_t)src * HC1 + hc);
  const float4 r4 = *(const float4*)(xr + (size_t)dst * HC1 + hc);
  const float4 a4 = *(const float4*)(att + hc);
  float p = a4.x * lrelu(l4.x + r4.x) + a4.y * lrelu(l4.y + r4.y)
          + a4.z * lrelu(l4.z + r4.z) + a4.w * lrelu(l4.w + r4.w);
  p += __shfl_xor(p, 1, 32);
  p += __shfl_xor(p, 2, 32);
  if ((lane & 3) == 0) {
    logits[(size_t)e * HEADS1 + h] = p;
    atomicMaxF(&maxv[(size_t)dst * HEADS1 + h], p);
  }
}

__global__ void gat_e1_expsum(const int* __restrict__ ei,
                              float* __restrict__ logits,
                              const float* __restrict__ maxv,
                              float* __restrict__ denom) {
  const long long i = (long long)blockIdx.x * blockDim.x + threadIdx.x;
  if (i >= (long long)NTOT * HEADS1) return;
  const int e = (int)(i >> 3);
  const int h = (int)(i & 7);
  const int dst = (e < NEDGES) ? ei[NEDGES + e] : e - NEDGES;
  const float a = __expf(logits[i] - maxv[(size_t)dst * HEADS1 + h]);
  logits[i] = a;                    // cache exp value for the aggregate pass
  atomicAdd(&denom[(size_t)dst * HEADS1 + h], a);
}

__global__ void gat_e1_aggregate(const int* __restrict__ ei,
                                 const float* __restrict__ xl,
                                 const float* __restrict__ logits,
                                 const float* __restrict__ denom,
                                 float* __restrict__ agg) {
  const int lane = threadIdx.x & 31;
  const int e = blockIdx.x * (blockDim.x >> 5) + (threadIdx.x >> 5);
  if (e >= NTOT) return;
  int src, dst; edge_endpoints(ei, e, src, dst);
  const int h  = lane >> 2;
  const int hc = h * CH + (lane & 3) * 4;
  const float a = logits[(size_t)e * HEADS1 + h];
  const float d = denom[(size_t)dst * HEADS1 + h];
  const float alpha = a / (d + 1e-16f);
  const float4 l4 = *(const float4*)(xl + (size_t)src * HC1 + hc);
  float* o = agg + (size_t)dst * HC1 + hc;
  atomicAdd(o + 0, l4.x * alpha);
  atomicAdd(o + 1, l4.y * alpha);
  atomicAdd(o + 2, l4.z * alpha);
  atomicAdd(o + 3, l4.w * alpha);
}

// buf[i] = elu(buf[i] + bias[i % bmod])
__global__ void gat_bias_elu(const float* __restrict__ bias,
                             float* __restrict__ buf, long long n, int bmod) {
  const long long i = (long long)blockIdx.x * blockDim.x + threadIdx.x;
  if (i >= n) return;
  buf[i] = eluf(buf[i] + bias[(int)(i % bmod)]);
}

// ---------------------------------------------------------------------------
// Layer 2 edge passes (H=1, C=16): one thread per edge
// ---------------------------------------------------------------------------
__global__ void gat_e2_logits_max(const int* __restrict__ ei,
                                  const float* __restrict__ xl,
                                  const float* __restrict__ xr,
                                  const float* __restrict__ att,
                                  float* __restrict__ logits,
                                  float* __restrict__ maxv) {
  const int e = blockIdx.x * blockDim.x + threadIdx.x;
  if (e >= NTOT) return;
  int src, dst; edge_endpoints(ei, e, src, dst);
  float p = 0.f;
#pragma unroll
  for (int j = 0; j < 4; ++j) {
    const float4 l4 = *(const float4*)(xl + (size_t)src * CH + 4 * j);
    const float4 r4 = *(const float4*)(xr + (size_t)dst * CH + 4 * j);
    const float4 a4 = *(const float4*)(att + 4 * j);
    p += a4.x * lrelu(l4.x + r4.x) + a4.y * lrelu(l4.y + r4.y)
       + a4.z * lrelu(l4.z + r4.z) + a4.w * lrelu(l4.w + r4.w);
  }
  logits[e] = p;
  atomicMaxF(&maxv[dst], p);
}

__global__ void gat_e2_expsum(const int* __restrict__ ei,
                              float* __restrict__ logits,
                              const float* __restrict__ maxv,
                              float* __restrict__ denom) {
  const int e = blockIdx.x * blockDim.x + threadIdx.x;
  if (e >= NTOT) return;
  const int dst = (e < NEDGES) ? ei[NEDGES + e] : e - NEDGES;
  const float a = __expf(logits[e] - maxv[dst]);
  logits[e] = a;
  atomicAdd(&denom[dst], a);
}

__global__ void gat_e2_aggregate(const int* __restrict__ ei,
                                 const float* __restrict__ xl,
                                 const float* __restrict__ logits,
                                 const float* __restrict__ denom,
                                 float* __restrict__ agg) {
  const int e = blockIdx.x * blockDim.x + threadIdx.x;
  if (e >= NTOT) return;
  int src, dst; edge_endpoints(ei, e, src, dst);
  const float alpha = logits[e] / (denom[dst] + 1e-16f);
  float* o = agg + (size_t)dst * CH;
#pragma unroll
  for (int j = 0; j < 4; ++j) {
    const float4 l4 = *(const float4*)(xl + (size_t)src * CH + 4 * j);
    atomicAdd(o + 4 * j + 0, l4.x * alpha);
    atomicAdd(o + 4 * j + 1, l4.y * alpha);
    atomicAdd(o + 4 * j + 2, l4.z * alpha);
    atomicAdd(o + 4 * j + 3, l4.w * alpha);
  }
}

// ---------------------------------------------------------------------------
// Global mean pool (sum + count) and fused divide + 3-layer MLP
// ---------------------------------------------------------------------------
__global__ void gat_pool_sum(const int* __restrict__ batch,
                             const float* __restrict__ h2,
                             float* __restrict__ pooled,
                             float* __restrict__ cnt) {
  const int n = blockIdx.x * blockDim.x + threadIdx.x;
  if (n >= NNODES) return;
  const int b = batch[n];
#pragma unroll
  for (int c = 0; c < CH; ++c)
    atomicAdd(&pooled[(size_t)b * CH + c], h2[(size_t)n * CH + c]);
  atomicAdd(&cnt[b], 1.f);
}

__global__ void gat_mlp(const float* __restrict__ pooled,
                        const float* __restrict__ cnt,
                        const float* __restrict__ w1, const float* __restrict__ b1,
                        const float* __restrict__ w2, const float* __restrict__ b2,
                        const float* __restrict__ w3, const float* __restrict__ b3,
                        float* __restrict__ out) {
  const int b = blockIdx.x * blockDim.x + threadIdx.x;
  if (b >= NB) return;
  const float inv = 1.f / fmaxf(cnt[b], 1.f);
  float h[CH];
#pragma unroll
  for (int c = 0; c < CH; ++c) h[c] = pooled[(size_t)b * CH + c] * inv;
  float t1[2 * CH];
#pragma unroll
  for (int o = 0; o < 2 * CH; ++o) {
    float s = b1[o];
    for (int c = 0; c < CH; ++c) s += h[c] * w1[c * (2 * CH) + o];
    t1[o] = eluf(s);
  }
  float t2[CH];
#pragma unroll
  for (int o = 0; o < CH; ++o) {
    float s = b2[o];
    for (int c = 0; c < 2 * CH; ++c) s += t1[c] * w2[c * CH + o];
    t2[o] = eluf(s);
  }
#pragma unroll
  for (int o = 0; o < OUTC; ++o) {
    float s = b3[o];
    for (int c = 0; c < CH; ++c) s += t2[c] * w3[c * OUTC + o];
    out[(size_t)b * OUTC + o] = s;
  }
}

// ---------------------------------------------------------------------------
extern "C" void kernel_launch(void* const* d_in, const int* in_sizes, int n_in,
                              void* d_out, int out_size, void* d_ws, size_t ws_size,
                              hipStream_t stream) {
  const float* x     = (const float*)d_in[0];
  const int*   ei    = (const int*)d_in[1];
  const int*   batch = (const int*)d_in[2];
  const float* wl1   = (const float*)d_in[3];
  const float* bl1   = (const float*)d_in[4];
  const float* wr1   = (const float*)d_in[5];
  const float* br1   = (const float*)d_in[6];
  const float* att1  = (const float*)d_in[7];
  const float* bias1 = (const float*)d_in[8];
  const float* wl2   = (const float*)d_in[9];
  const float* bl2   = (const float*)d_in[10];
  const float* wr2   = (const float*)d_in[11];
  const float* br2   = (const float*)d_in[12];
  const float* att2  = (const float*)d_in[13];
  const float* bias2 = (const float*)d_in[14];
  const float* w_m1  = (const float*)d_in[15];
  const float* b_m1  = (const float*)d_in[16];
  const float* w_m2  = (const float*)d_in[17];
  const float* b_m2  = (const float*)d_in[18];
  const float* w_m3  = (const float*)d_in[19];
  const float* b_m3  = (const float*)d_in[20];
  float* out = (float*)d_out;

  // workspace carve-out
  char* ws = (char*)d_ws;
  size_t cur = 0;
  auto carve = [&](size_t bytes) -> float* {
    float* p = (float*)(ws + cur);
    cur = (cur + bytes + 255) & ~(size_t)255;
    return p;
  };
  float* xl1    = carve((size_t)NNODES * HC1 * 4);
  float* xr1    = carve((size_t)NNODES * HC1 * 4);
  float* agg1   = carve((size_t)NNODES * HC1 * 4);   // becomes h1 after bias+elu
  float* log1   = carve((size_t)NTOT * HEADS1 * 4);
  float* max1   = carve((size_t)NNODES * HEADS1 * 4);
  float* den1   = carve((size_t)NNODES * HEADS1 * 4);
  float* xl2    = carve((size_t)NNODES * CH * 4);
  float* xr2    = carve((size_t)NNODES * CH * 4);
  float* agg2   = carve((size_t)NNODES * CH * 4);    // becomes h2
  float* log2   = carve((size_t)NTOT * 4);
  float* max2   = carve((size_t)NNODES * 4);
  float* den2   = carve((size_t)NNODES * 4);
  float* pooled = carve((size_t)NB * CH * 4);
  float* cnt    = carve((size_t)NB * 4);

  const int TB = 256;
  auto cdiv = [](long long a, long long b) { return (int)((a + b - 1) / b); };

  // init accumulators
  hipMemsetAsync(agg1, 0, (size_t)NNODES * HC1 * 4, stream);
  hipMemsetAsync(den1, 0, (size_t)NNODES * HEADS1 * 4, stream);
  hipMemsetAsync(agg2, 0, (size_t)NNODES * CH * 4, stream);
  hipMemsetAsync(den2, 0, (size_t)NNODES * 4, stream);
  hipMemsetAsync(pooled, 0, (size_t)NB * CH * 4, stream);
  hipMemsetAsync(cnt, 0, (size_t)NB * 4, stream);
  gat_fill_f32<<<cdiv(NNODES * HEADS1, TB), TB, 0, stream>>>(max1, -INFINITY, NNODES * HEADS1);
  gat_fill_f32<<<cdiv(NNODES, TB), TB, 0, stream>>>(max2, -INFINITY, NNODES);

  // ---- layer 1 ----
  {
    const int waves = (NNODES / 16) * (HC1 / 16);
    const int blocks = cdiv((long long)waves * 32, TB);
    gat_gemm_bias_wmma<<<blocks, TB, 0, stream>>>(x, wl1, bl1, xl1, NNODES, INCH, HC1);
    gat_gemm_bias_wmma<<<blocks, TB, 0, stream>>>(x, wr1, br1, xr1, NNODES, INCH, HC1);
  }
  {
    const int blocks = cdiv((long long)NTOT * 32, TB);
    gat_e1_logits_max<<<blocks, TB, 0, stream>>>(ei, xl1, xr1, att1, log1, max1);
    gat_e1_expsum<<<cdiv((long long)NTOT * HEADS1, TB), TB, 0, stream>>>(ei, log1, max1, den1);
    gat_e1_aggregate<<<blocks, TB, 0, stream>>>(ei, xl1, log1, den1, agg1);
    gat_bias_elu<<<cdiv((long long)NNODES * HC1, TB), TB, 0, stream>>>(
        bias1, agg1, (long long)NNODES * HC1, HC1);
  }

  // ---- layer 2 ----
  {
    const int waves = (NNODES / 16) * (CH / 16);
    const int blocks = cdiv((long long)waves * 32, TB);
    gat_gemm_bias_wmma<<<blocks, TB, 0, stream>>>(agg1, wl2, bl2, xl2, NNODES, HC1, CH);
    gat_gemm_bias_wmma<<<blocks, TB, 0, stream>>>(agg1, wr2, br2, xr2, NNODES, HC1, CH);
  }
  {
    const int blocks = cdiv(NTOT, TB);
    gat_e2_logits_max<<<blocks, TB, 0, stream>>>(ei, xl2, xr2, att2, log2, max2);
    gat_e2_expsum<<<blocks, TB, 0, stream>>>(ei, log2, max2, den2);
    gat_e2_aggregate<<<blocks, TB, 0, stream>>>(ei, xl2, log2, den2, agg2);
    gat_bias_elu<<<cdiv((long long)NNODES * CH, TB), TB, 0, stream>>>(
        bias2, agg2, (long long)NNODES * CH, CH);
  }

  // ---- pool + MLP ----
  gat_pool_sum<<<cdiv(NNODES, TB), TB, 0, stream>>>(batch, agg2, pooled, cnt);
  gat_mlp<<<cdiv(NB, TB), TB, 0, stream>>>(pooled, cnt, w_m1, b_m1, w_m2, b_m2,
                                           w_m3, b_m3, out);
  (void)in_sizes; (void)n_in; (void)out_size; (void)ws_size;
}